// QuantizationLayer_69217692942365
// MI455X (gfx1250) — compile-verified
//
#include <hip/hip_runtime.h>
#include <hip/hip_bf16.h>
#include <stdint.h>

// Problem constants (fixed by the reference module)
#define HH  256
#define WWD 256
#define SSL 16        // time slices S
#define BBN 8         // batches B
#define MMC 4         // combiner channels M
#define HWP 65536     // H*W

typedef __attribute__((ext_vector_type(2))) float v2f;
typedef __attribute__((ext_vector_type(8))) float v8f;

// Tridiagonal band matrix entry: A16[m][k] = 1 if |m-k|<=1 else 0
__device__ __forceinline__ float band(int m, int k) {
  return ((unsigned)(m - k + 1) <= 2u) ? 1.0f : 0.0f;
}

// ---------------------------------------------------------------- init
__global__ void k_init(uint32_t* dilw, int* smalls, int nwords) {
  int stride = gridDim.x * blockDim.x;
  int tid = blockIdx.x * blockDim.x + threadIdx.x;
  for (int i = tid; i < nwords; i += stride) dilw[i] = 0u;
  if (tid < 184) smalls[tid] = 0;  // tmax[8], npix[8], cmax[8], csum[5*32]
}

// ---------------------------------------------------------------- per-batch t_max
__global__ void k_tmax(const float* __restrict__ ev, int n, int* tmaxbits) {
  __shared__ int smax[BBN];
  if (threadIdx.x < BBN) smax[threadIdx.x] = 0;
  __syncthreads();
  int i = blockIdx.x * blockDim.x + threadIdx.x;
  if (i < n) {
    float t = ev[(size_t)i * 5 + 2];
    int b = (int)ev[(size_t)i * 5 + 4];
    atomicMax(&smax[b & 7], __float_as_int(t));  // t >= 0: int-max == float-max
  }
  __syncthreads();
  if (threadIdx.x < BBN) atomicMax(&tmaxbits[threadIdx.x], smax[threadIdx.x]);
}

// ---------------------------------------------------------------- event scatter
__global__ void k_scatter(const float* __restrict__ ev, int n,
                          const int* __restrict__ tmaxbits, uint8_t* dil) {
  int i = blockIdx.x * blockDim.x + threadIdx.x;
  if (i >= n) return;
  const float* e = ev + (size_t)i * 5;
  __builtin_prefetch(e + 5 * 512, 0, 0);   // global_prefetch_b8, stream ahead
  int x = (int)e[0];
  int y = (int)e[1];
  float t = e[2];
  int b = ((int)e[4]) & 7;
  float tm = __int_as_float(tmaxbits[b]);
  int ts = (int)((t / tm) * (float)SSL);
  ts = ts > SSL - 1 ? SSL - 1 : ts;
  dil[(size_t)(((b * SSL + ts) * HH + y) * WWD + x)] = 1;  // racing same-value stores OK
}

// ---------------------------------------------------------------- scan seed (mix[0])
__global__ void k_seed(const uint8_t* __restrict__ dil, float* prev, int* conc) {
  int idx = blockIdx.x * blockDim.x + threadIdx.x;  // over B*HW
  int b = idx >> 16;
  int pix = idx & (HWP - 1);
  float m0 = (float)dil[(size_t)(b * SSL) * HWP + pix];
  prev[idx] = m0;
  conc[idx] = (m0 > 0.f) ? 1 : 0;
}

// ---------------------------------------------------------------- vertical band matmul:
// V = A16 * Z (column 3-sum), Z computed on the fly = dil_i*w1 + prev*w2.
// One wave per 16x16 tile; halo rows preloaded into C rows 0 / 15.
__global__ void k_vert(const uint8_t* __restrict__ dil, const float* __restrict__ prev,
                       float* __restrict__ vimg, int slice, float w1, float w2) {
  int lane = threadIdx.x & 31;
  int tile = blockIdx.x * (blockDim.x >> 5) + (threadIdx.x >> 5);  // 0..2047
  int b = tile >> 8;
  int r = tile & 255;
  int y0 = (r >> 4) << 4;
  int x0 = (r & 15) << 4;
  const uint8_t* ds = dil + (size_t)(b * SSL + slice) * HWP;
  const float* pv = prev + (size_t)b * HWP;
  int col = lane & 15, half = lane >> 4;

  auto pre = [&](int y, int x) -> float {
    int o = y * WWD + x;
    return (float)ds[o] * w1 + pv[o] * w2;
  };

  // C/D layout: VGPR j -> rows M=j (lanes 0-15) and M=j+8 (lanes 16-31), N=lane%16.
  v8f c = {0.f, 0.f, 0.f, 0.f, 0.f, 0.f, 0.f, 0.f};
  if (half == 0) { if (y0 > 0)        c[0] = pre(y0 - 1,  x0 + col); }  // row -1
  else           { if (y0 + 16 < HH)  c[7] = pre(y0 + 16, x0 + col); }  // row 16

#pragma unroll
  for (int k = 0; k < 4; ++k) {                 // chain K in slices of 4
    int ka = 4 * k + (half ? 2 : 0);
    v2f a, bb;
    a.x = band(col, ka);                        // A operand: M = lane&15
    a.y = band(col, ka + 1);
    bb.x = pre(y0 + ka,     x0 + col);          // B operand: rows K across lanes
    bb.y = pre(y0 + ka + 1, x0 + col);
    c = __builtin_amdgcn_wmma_f32_16x16x4_f32(false, a, false, bb, (short)0, c,
                                              false, false);
  }

  float* vout = vimg + (size_t)b * HWP;
  int rowb = y0 + (half ? 8 : 0);
#pragma unroll
  for (int j = 0; j < 8; ++j) vout[(rowb + j) * WWD + x0 + col] = c[j];
}

// ---------------------------------------------------------------- horizontal band matmul:
// Out = V * A16 (row 3-sum), then cur = (Out - center)/8 - 0.25; prev <- cur; conc += cur>0
__global__ void k_horz(const uint8_t* __restrict__ dil, const float* __restrict__ vimg,
                       float* __restrict__ prev, int* __restrict__ conc,
                       int slice, float w1, float w2) {
  int lane = threadIdx.x & 31;
  int tile = blockIdx.x * (blockDim.x >> 5) + (threadIdx.x >> 5);
  int b = tile >> 8;
  int r = tile & 255;
  int y0 = (r >> 4) << 4;
  int x0 = (r & 15) << 4;
  const float* V = vimg + (size_t)b * HWP;
  int col = lane & 15, half = lane >> 4;
  int mbase = half ? 8 : 0;

  v8f c = {0.f, 0.f, 0.f, 0.f, 0.f, 0.f, 0.f, 0.f};
  if (col == 0 && x0 > 0) {                      // column -1 halo -> N=0 lanes
#pragma unroll
    for (int j = 0; j < 8; ++j) c[j] = V[(y0 + mbase + j) * WWD + x0 - 1];
  }
  if (col == 15 && x0 + 16 < WWD) {              // column 16 halo -> N=15 lanes
#pragma unroll
    for (int j = 0; j < 8; ++j) c[j] = V[(y0 + mbase + j) * WWD + x0 + 16];
  }

#pragma unroll
  for (int k = 0; k < 4; ++k) {
    int ka = 4 * k + (half ? 2 : 0);
    v2f a, bb;
    a.x = V[(y0 + col) * WWD + x0 + ka];         // A operand: M = lane&15, K slices
    a.y = V[(y0 + col) * WWD + x0 + ka + 1];
    bb.x = band(ka, col);                        // B operand: A16 rows K, cols N
    bb.y = band(ka + 1, col);
    c = __builtin_amdgcn_wmma_f32_16x16x4_f32(false, a, false, bb, (short)0, c,
                                              false, false);
  }

  const uint8_t* ds = dil + (size_t)(b * SSL + slice) * HWP;
  float* pv = prev + (size_t)b * HWP;
  int* cc = conc + (size_t)b * HWP;
#pragma unroll
  for (int j = 0; j < 8; ++j) {
    int o = (y0 + mbase + j) * WWD + x0 + col;
    float p = (float)ds[o] * w1 + pv[o] * w2;    // recompute center (prev not yet clobbered)
    float cur = (c[j] - p) * 0.125f - 0.25f;
    pv[o] = cur;
    if (cur > 0.f) cc[o] += 1;
  }
}

// ---------------------------------------------------------------- per-batch count / max
__global__ void k_reduce(const int* __restrict__ conc, int* npix, int* cmax) {
  __shared__ int ssum[256];
  __shared__ int smx[256];
  int idx = blockIdx.x * 256 + threadIdx.x;      // over B*HW (each block: one batch)
  int b = idx >> 16;
  int cv = conc[idx];
  ssum[threadIdx.x] = (cv > 0) ? 1 : 0;
  smx[threadIdx.x] = cv;
  __syncthreads();
  for (int s = 128; s > 0; s >>= 1) {
    if (threadIdx.x < s) {
      ssum[threadIdx.x] += ssum[threadIdx.x + s];
      smx[threadIdx.x] = max(smx[threadIdx.x], smx[threadIdx.x + s]);
    }
    __syncthreads();
  }
  if (threadIdx.x == 0) {
    atomicAdd(&npix[b], ssum[0]);
    atomicMax(&cmax[b], smx[0]);
  }
}

// ---------------------------------------------------------------- combiner stage 0
__global__ void k_comb0(const uint8_t* __restrict__ dil, const int* __restrict__ conc,
                        uint8_t* comb, int* csum0, int i0, int i1, int i2, int i3) {
  __shared__ int ssum[256];
  int gid = blockIdx.x * 256 + threadIdx.x;      // over B*M*HW
  int pix = gid & (HWP - 1);
  int bm = gid >> 16;
  int b = bm >> 2, m = bm & 3;
  int idxr = (m == 0) ? i0 : (m == 1) ? i1 : (m == 2) ? i2 : i3;
  int rl = (conc[b * HWP + pix] > 0) ? 1 : 0;
  uint8_t v = (uint8_t)(dil[(size_t)(b * SSL + idxr) * HWP + pix] & rl);
  comb[(size_t)bm * HWP + pix] = v;
  ssum[threadIdx.x] = v;
  __syncthreads();
  for (int s = 128; s > 0; s >>= 1) {
    if (threadIdx.x < s) ssum[threadIdx.x] += ssum[threadIdx.x + s];
    __syncthreads();
  }
  if (threadIdx.x == 0) atomicAdd(&csum0[bm], ssum[0]);
}

// ---------------------------------------------------------------- combiner stage cnt
__global__ void k_combk(const uint8_t* __restrict__ dil, const int* __restrict__ conc,
                        uint8_t* comb, const int* __restrict__ csum_prev, int* csum_cur,
                        const int* __restrict__ npix, int j0, int j1, int j2, int j3) {
  __shared__ int ssum[256];
  int gid = blockIdx.x * 256 + threadIdx.x;
  int pix = gid & (HWP - 1);
  int bm = gid >> 16;
  int b = bm >> 2, m = bm & 3;
  int jm = (m == 0) ? j0 : (m == 1) ? j1 : (m == 2) ? j2 : j3;
  int need = csum_prev[bm] < (npix[b] >> 4);     // min_pixel_required = npix // S
  uint8_t v = comb[(size_t)bm * HWP + pix];
  if (need) {
    int rl = (conc[b * HWP + pix] > 0) ? 1 : 0;
    v = (uint8_t)(v | (dil[(size_t)(b * SSL + jm) * HWP + pix] & rl));
    comb[(size_t)bm * HWP + pix] = v;
  }
  ssum[threadIdx.x] = v;
  __syncthreads();
  for (int s = 128; s > 0; s >>= 1) {
    if (threadIdx.x < s) ssum[threadIdx.x] += ssum[threadIdx.x + s];
    __syncthreads();
  }
  if (threadIdx.x == 0) atomicAdd(&csum_cur[bm], ssum[0]);
}

// ---------------------------------------------------------------- final output (B,5,H,W)
__global__ void k_out(const int* __restrict__ conc, const uint8_t* __restrict__ comb,
                      const int* __restrict__ cmax, float* __restrict__ out) {
  int idx = blockIdx.x * 256 + threadIdx.x;      // over B*HW
  int b = idx >> 16;
  int pix = idx & (HWP - 1);
  float cm = (float)cmax[b];
  out[(size_t)(b * 5) * HWP + pix] = (float)conc[idx] / cm;
#pragma unroll
  for (int m = 0; m < 4; ++m)
    out[(size_t)(b * 5 + 1 + m) * HWP + pix] =
        comb[(size_t)(b * 4 + m) * HWP + pix] ? 1.f : 0.f;
}

// ================================================================ launcher
extern "C" void kernel_launch(void* const* d_in, const int* in_sizes, int n_in,
                              void* d_out, int out_size, void* d_ws, size_t ws_size,
                              hipStream_t stream) {
  const float* ev = (const float*)d_in[0];
  int n = in_sizes[0] / 5;

  // workspace carve-up (~16.8 MB)
  char* ws = (char*)d_ws;
  size_t off = 0;
  uint8_t* dil  = (uint8_t*)(ws + off); off += (size_t)BBN * SSL * HWP;     // 8 MiB
  float*   prev = (float*)(ws + off);   off += (size_t)BBN * HWP * 4;       // 2 MiB
  float*   vimg = (float*)(ws + off);   off += (size_t)BBN * HWP * 4;       // 2 MiB
  int*     conc = (int*)(ws + off);     off += (size_t)BBN * HWP * 4;       // 2 MiB
  uint8_t* comb = (uint8_t*)(ws + off); off += (size_t)BBN * MMC * HWP;     // 2 MiB
  int* smalls   = (int*)(ws + off);
  int* tmaxbits = smalls;        // [8]
  int* npix     = smalls + 8;    // [8]
  int* cmax     = smalls + 16;   // [8]
  int* csum     = smalls + 24;   // [5][32]

  k_init<<<2048, 256, 0, stream>>>((uint32_t*)dil, smalls, BBN * SSL * HWP / 4);
  k_tmax<<<(n + 255) / 256, 256, 0, stream>>>(ev, n, tmaxbits);
  k_scatter<<<(n + 255) / 256, 256, 0, stream>>>(ev, n, tmaxbits, dil);
  k_seed<<<BBN * HWP / 256, 256, 0, stream>>>(dil, prev, conc);

  for (int i = 1; i < SSL; ++i) {
    float w1 = 0.5f + (float)(SSL - i) / (float)SSL;
    float w2 = (float)i / (float)SSL;
    k_vert<<<256, 256, 0, stream>>>(dil, prev, vimg, i, w1, w2);
    k_horz<<<256, 256, 0, stream>>>(dil, vimg, prev, conc, i, w1, w2);
  }

  k_reduce<<<BBN * HWP / 256, 256, 0, stream>>>(conc, npix, cmax);

  const int RI[4] = {12, 3, 7, 0};
  k_comb0<<<BBN * MMC * HWP / 256, 256, 0, stream>>>(dil, conc, comb, csum,
                                                     RI[0], RI[1], RI[2], RI[3]);
  for (int cnt = 1; cnt <= 4; ++cnt) {
    int j[4];
    for (int m = 0; m < 4; ++m) {
      int ir = RI[m];
      j[m] = (ir < MMC / 2) ? (ir + cnt) % SSL : (((ir - cnt) % SSL) + SSL) % SSL;
    }
    k_combk<<<BBN * MMC * HWP / 256, 256, 0, stream>>>(
        dil, conc, comb, csum + (cnt - 1) * 32, csum + cnt * 32, npix,
        j[0], j[1], j[2], j[3]);
  }

  k_out<<<BBN * HWP / 256, 256, 0, stream>>>(conc, comb, cmax, (float*)d_out);
}